// AutomatonPELayer_7516192768713
// MI455X (gfx1250) — compile-verified
//
#include <hip/hip_runtime.h>
#include <hip/hip_bf16.h>

// Problem constants from the reference (compile-time; sentence_len input == SEQ).
#define K_STATES 16
#define SEQ      512
#define EMB      512

typedef __attribute__((ext_vector_type(2))) float v2f;
typedef __attribute__((ext_vector_type(8))) float v8f;

// ---------------------------------------------------------------------------
// Kernel 1: backward recurrence  x_{n-1} = v0 ; x_a = v0 + T * x_{a+1}
// One wave32. Lane s (and its mirror s+16) owns state component s; row s of T
// is held in registers; the current state vector is broadcast with __shfl.
// Writes S[SEQ][16] fp32 into workspace.
// ---------------------------------------------------------------------------
__global__ __launch_bounds__(32) void automaton_recur_kernel(
    const float* __restrict__ pos_initial,   // [16]
    const float* __restrict__ T,             // [16][16] row-major
    float* __restrict__ S)                   // [SEQ][16] out
{
    const int lane = threadIdx.x;
    const int s    = lane & 15;              // lanes 16..31 mirror 0..15

    const float v0 = pos_initial[s];

    float Trow[K_STATES];
#pragma unroll
    for (int t = 0; t < K_STATES; ++t)
        Trow[t] = T[s * K_STATES + t];

    float x = v0;                            // x_{n-1}
    if (lane < K_STATES)
        S[(SEQ - 1) * K_STATES + s] = x;

    for (int a = SEQ - 2; a >= 0; --a) {
        float acc = v0;
#pragma unroll
        for (int t = 0; t < K_STATES; ++t)
            acc = fmaf(Trow[t], __shfl(x, t, 32), acc);
        x = acc;
        if (lane < K_STATES)
            S[a * K_STATES + s] = x;
    }
}

// ---------------------------------------------------------------------------
// Kernel 2: pe = S[512,16] @ W^T[16,512] + b  via V_WMMA_F32_16X16X4_F32.
// One wave32 per 16x16 output tile; K=16 -> 4 chained WMMA ops (K chunks of 4).
//
// Fragment layouts (CDNA5 ISA 7.12.2, 32-bit data, wave32):
//   A 16x4 : lane l -> M = l&15 ; VGPR0/1 = K = (l>>4)*2 + {0,1}  (per chunk)
//   B 4x16 : lane l -> N = l&15 ; VGPR0/1 = K = (l>>4)*2 + {0,1}
//   C/D    : lane l -> N = l&15 ; VGPR r  = M = r + 8*(l>>4)
// ---------------------------------------------------------------------------
__global__ __launch_bounds__(32) void automaton_pe_wmma_kernel(
    const float* __restrict__ S,             // [SEQ][16]
    const float* __restrict__ W,             // [EMB][16] (row-major, W[e][s])
    const float* __restrict__ bias,          // [EMB]
    float* __restrict__ out)                 // [SEQ][EMB]
{
    const int tile = blockIdx.x;
    const int ti   = tile >> 5;              // 0..31 : row tile  (SEQ/16)
    const int tj   = tile & 31;              // 0..31 : col tile  (EMB/16)
    const int i0   = ti * 16;
    const int j0   = tj * 16;

    const int lane = threadIdx.x;
    const int m    = lane & 15;              // M for A, N for B/C/D
    const int half = lane >> 4;

    // Row pointers: A-lane reads row (i0+m) of S, B-lane reads row (j0+m) of W
    // (W row e holds W[e][0..15] == column e of W^T).
    const float* __restrict__ Srow = S + (i0 + m) * K_STATES;
    const float* __restrict__ Wrow = W + (j0 + m) * K_STATES;

    v8f acc = {};
#pragma unroll
    for (int c = 0; c < 4; ++c) {
        const int kb = c * 4 + half * 2;     // K base for this lane-half
        v2f afrag, bfrag;
        afrag.x = Srow[kb];
        afrag.y = Srow[kb + 1];
        bfrag.x = Wrow[kb];
        bfrag.y = Wrow[kb + 1];
        // D = A(16x4) * B(4x16) + C   — fp32 in, fp32 accumulate
        acc = __builtin_amdgcn_wmma_f32_16x16x4_f32(
            /*neg_a=*/false, afrag,
            /*neg_b=*/false, bfrag,
            /*c_mod=*/(short)0, acc,
            /*reuse_a=*/false, /*reuse_b=*/false);
    }

    const float bv = bias[j0 + m];
#pragma unroll
    for (int r = 0; r < 8; ++r) {
        const int row = i0 + r + 8 * half;
        out[row * EMB + j0 + m] = acc[r] + bv;
    }
}

// ---------------------------------------------------------------------------
// Launch
// ---------------------------------------------------------------------------
extern "C" void kernel_launch(void* const* d_in, const int* in_sizes, int n_in,
                              void* d_out, int out_size, void* d_ws, size_t ws_size,
                              hipStream_t stream) {
    const float* pos_initial = (const float*)d_in[0];   // [16,1]
    const float* T           = (const float*)d_in[1];   // [16,16]
    const float* W           = (const float*)d_in[2];   // [512,16]
    const float* b           = (const float*)d_in[3];   // [512]
    // d_in[4] = sentence_len scalar == SEQ (compile-time constant here)

    float* S   = (float*)d_ws;    // [SEQ][16] fp32 = 32 KB scratch
    float* out = (float*)d_out;   // [SEQ][EMB] fp32

    automaton_recur_kernel<<<1, 32, 0, stream>>>(pos_initial, T, S);

    const int tiles = (SEQ / 16) * (EMB / 16);  // 1024
    automaton_pe_wmma_kernel<<<tiles, 32, 0, stream>>>(S, W, b, out);
}